// EEGPreprocessor_26749056319870
// MI455X (gfx1250) — compile-verified
//
#include <hip/hip_runtime.h>

typedef __attribute__((ext_vector_type(2))) float v2f;
typedef __attribute__((ext_vector_type(4))) float v4f;
typedef __attribute__((ext_vector_type(8))) float v8f;

#define BATCH 64
#define CH    128
#define TLEN  10000
#define TVEC  (TLEN / 4)           // 2500 float4 per row
#define TILES (TLEN / 16)          // 625 time-tiles of 16
#define UNITS (BATCH * TILES)      // 40000 wave-units for kernel 1

#if __has_builtin(__builtin_amdgcn_wmma_f32_16x16x4_f32)
#define HAVE_WMMA_F32X4 1
#else
#define HAVE_WMMA_F32X4 0
#endif

// ---------------------------------------------------------------------------
// Kernel 1: mu[b,t] = mean over 128 channels of x[b,c,t].
// One wave computes 16 consecutive t values. The channel reduction is done as
// D = A(16x4, all 1/128) x B(4x16 tile of x) + C accumulated over 32 WMMAs.
// ---------------------------------------------------------------------------
__global__ __launch_bounds__(256) void chan_mean_wmma(const float* __restrict__ x,
                                                      float* __restrict__ mu) {
    const int lane  = threadIdx.x & 31;
    const int wid   = (blockIdx.x << 3) + (threadIdx.x >> 5);  // 0..UNITS-1 (exact grid)
    const int b     = wid / TILES;
    const int t0    = (wid % TILES) << 4;
    const int t     = t0 + (lane & 15);
    const int khalf = (lane >> 4) << 1;                        // 0 for lanes 0-15, 2 for 16-31
    const float* base = x + (size_t)b * CH * TLEN + t;

#if HAVE_WMMA_F32X4
    v8f acc = {};
    v2f a;
    a.x = 0.0078125f;  // 1/128
    a.y = 0.0078125f;
    #pragma unroll 4
    for (int c0 = 0; c0 < CH; c0 += 4) {
        v2f bv;
        bv.x = base[(size_t)(c0 + khalf)     * TLEN];
        bv.y = base[(size_t)(c0 + khalf + 1) * TLEN];
        // D = A x B + C ; every row of D holds the same per-t partial mean.
        acc = __builtin_amdgcn_wmma_f32_16x16x4_f32(
            /*neg_a=*/false, a, /*neg_b=*/false, bv,
            /*c_mod=*/(short)0, acc, /*reuse_a=*/false, /*reuse_b=*/false);
    }
    float result = acc[0];                 // row M=0, n = lane (lanes 0-15)
#else
    float partial = 0.0f;
    #pragma unroll 4
    for (int c0 = 0; c0 < CH; c0 += 4) {
        partial += base[(size_t)(c0 + khalf)     * TLEN];
        partial += base[(size_t)(c0 + khalf + 1) * TLEN];
    }
    // lane L holds half the channels for t0+(L&15); other half is in lane L^16
    float result = (partial + __shfl_xor(partial, 16, 32)) * 0.0078125f;
#endif

    if (lane < 16)
        mu[(size_t)b * TLEN + t] = result;
}

// ---------------------------------------------------------------------------
// Kernel 2: per (b,c) row -> y = x - mu ; z = (y - mean_t(y)) / std_t(y).
// y is staged in LDS so x is read from HBM exactly once here.
// All global/LDS traffic is 128-bit vectorized (rows are 16B-aligned, 2500x16B).
// ---------------------------------------------------------------------------
__device__ __forceinline__ float waveReduceSum(float v) {
    #pragma unroll
    for (int off = 16; off > 0; off >>= 1)
        v += __shfl_down(v, off, 32);
    return v;
}

__global__ __launch_bounds__(256) void zscore_rows(const float* __restrict__ x,
                                                   const float* __restrict__ mu,
                                                   float* __restrict__ out) {
    __shared__ __align__(16) float yb[TLEN];   // 40000 B staged y
    __shared__ float red[8];
    __shared__ float stat[2];                  // [0]=mean, [1]=inv_std

    const int row = blockIdx.x;                // b*128 + c
    const int b   = row >> 7;
    const int tid = threadIdx.x;
    const v4f* xr4 = (const v4f*)(x  + (size_t)row * TLEN);
    const v4f* mr4 = (const v4f*)(mu + (size_t)b   * TLEN);
    v4f* orow4     = (v4f*)(out + (size_t)row * TLEN);
    v4f* yb4       = (v4f*)yb;

    // Phase A: y = x - mu, stage in LDS (b128), accumulate sum
    float lsum = 0.0f;
    for (int i = tid; i < TVEC; i += 256) {
        v4f xv = xr4[i];
        v4f mv = mr4[i];
        v4f yv = xv - mv;
        yb4[i] = yv;
        lsum += (yv.x + yv.y) + (yv.z + yv.w);
    }
    lsum = waveReduceSum(lsum);
    if ((tid & 31) == 0) red[tid >> 5] = lsum;
    __syncthreads();
    if (tid == 0) {
        float s = red[0] + red[1] + red[2] + red[3]
                + red[4] + red[5] + red[6] + red[7];
        stat[0] = s * (1.0f / TLEN);
    }
    __syncthreads();

    // Phase B: exact centered variance from LDS
    const float m = stat[0];
    float lvar = 0.0f;
    for (int i = tid; i < TVEC; i += 256) {
        v4f yv = yb4[i];
        float dx = yv.x - m, dy = yv.y - m, dz = yv.z - m, dw = yv.w - m;
        lvar += (dx * dx + dy * dy) + (dz * dz + dw * dw);
    }
    lvar = waveReduceSum(lvar);
    if ((tid & 31) == 0) red[tid >> 5] = lvar;
    __syncthreads();
    if (tid == 0) {
        float v = red[0] + red[1] + red[2] + red[3]
                + red[4] + red[5] + red[6] + red[7];
        float s = sqrtf(v * (1.0f / TLEN));
        stat[1] = (s == 0.0f) ? 1.0f : (1.0f / s);
    }
    __syncthreads();

    // Phase C: normalize, write with non-temporal stores (output never re-read)
    const float inv = stat[1];
    for (int i = tid; i < TVEC; i += 256) {
        v4f yv = yb4[i];
        v4f ov;
        ov.x = (yv.x - m) * inv;
        ov.y = (yv.y - m) * inv;
        ov.z = (yv.z - m) * inv;
        ov.w = (yv.w - m) * inv;
        __builtin_nontemporal_store(ov, &orow4[i]);
    }
}

extern "C" void kernel_launch(void* const* d_in, const int* in_sizes, int n_in,
                              void* d_out, int out_size, void* d_ws, size_t ws_size,
                              hipStream_t stream) {
    const float* x = (const float*)d_in[0];
    float* out = (float*)d_out;
    float* mu  = (float*)d_ws;   // 64*10000 floats = 2.56 MB scratch

    // Kernel 1: 40000 wave-units, 8 waves per 256-thread block -> 5000 blocks
    chan_mean_wmma<<<UNITS / 8, 256, 0, stream>>>(x, mu);
    // Kernel 2: one block per (batch, channel) row
    zscore_rows<<<BATCH * CH, 256, 0, stream>>>(x, mu, out);
}